// Warp3d_51427938402993
// MI455X (gfx1250) — compile-verified
//
#include <hip/hip_runtime.h>
#include <hip/hip_bf16.h>

// ---------------------------------------------------------------------------
// 3D trilinear warp (voxelmorph SpatialTransformer), MI455X / gfx1250.
// Memory-bound gather kernel: ~470MB traffic -> ~20us floor @ 23.3 TB/s.
//  - x (134MB) kept L2-resident (192MB L2): regular-temporal gathers
//  - w / out streamed with NT hints (no L2 pollution)
//  - flow staged via gfx1250 async-to-LDS loads (ASYNCcnt), lane-private
//    ping-pong buffers, no workgroup barrier needed
//  - batch index in blockIdx.y => scalar bases; every plane/channel base is
//    pinned to an SGPR pair so all accesses are saddr + 32-bit lane offset
//    (scale_offset), no per-lane 64-bit address math
// ---------------------------------------------------------------------------

#define DSZ   128
#define VOLN  (DSZ * DSZ * DSZ)      // 1 << 21
#define NB    8                      // b * n = 2 * 4
#define NC    2
#define TPB   256
#define IPT   4                      // items per thread (pipelined)

#ifndef __has_builtin
#define __has_builtin(x) 0
#endif

#if __has_builtin(__builtin_amdgcn_global_load_async_to_lds_b32) && \
    __has_builtin(__builtin_amdgcn_s_wait_asynccnt)
#define USE_ASYNC 1
#else
#define USE_ASYNC 0
#endif

// Pin a (wave-uniform) pointer into an SGPR pair; prevents the compiler from
// reassociating scalar plane offsets into per-lane 64-bit vector addresses.
#define PIN_SGPR(p) asm("" : "+s"(p))

#if USE_ASYNC
typedef __attribute__((address_space(1))) int gas_int;   // global int*
typedef __attribute__((address_space(3))) int las_int;   // LDS int*

__device__ __forceinline__ void async_load_b32(const float* g, float* l) {
  // Generic->AS casts via integer round-trip (LDS generic addr low 32 bits
  // are the LDS offset per CDNA5 aperture rules).
  __builtin_amdgcn_global_load_async_to_lds_b32(
      (gas_int*)(unsigned long long)g,
      (las_int*)(unsigned int)(unsigned long long)l,
      0, 0);
}
#endif

// Per-dimension prep: clamped lo/hi coords + validity-folded weights.
// Zero-padding: weight is zeroed for out-of-range taps, so the clamped
// address only needs to be in-bounds (gathered value is multiplied by 0).
__device__ __forceinline__ void prep_dim(float loc, float Wc[2], int Cc[2]) {
  const float f = floorf(loc);
  const float t = loc - f;
  const int lo = (int)f;
  Wc[0] = ((unsigned)lo < (unsigned)DSZ) ? (1.0f - t) : 0.0f;
  Wc[1] = ((unsigned)(lo + 1) < (unsigned)DSZ) ? t : 0.0f;
  int clo = lo < 0 ? 0 : (lo > DSZ - 1 ? DSZ - 1 : lo);   // -> v_med3_i32
  int chi = clo + 1;
  Cc[0] = clo;
  Cc[1] = chi > DSZ - 1 ? DSZ - 1 : chi;
}

// One output item: SGPR-resident channel bases, flat voxel v.
__device__ __forceinline__ void trilerp_item(const float* __restrict__ xb0,
                                             const float* __restrict__ xb1,
                                             float* __restrict__ ob0,
                                             float* __restrict__ ob1,
                                             int v,
                                             float w0, float w1, float w2) {
  const int i0 = v >> 14;
  const int i1 = (v >> 7) & (DSZ - 1);
  const int i2 = v & (DSZ - 1);

  float W0c[2], W1c[2], W2c[2];
  int C0[2], C1[2], C2[2];
  prep_dim((float)i0 + w0, W0c, C0);
  prep_dim((float)i1 + w1, W1c, C1);
  prep_dim((float)i2 + w2, W2c, C2);

  float acc0 = 0.0f, acc1 = 0.0f;
#pragma unroll
  for (int di = 0; di < 2; ++di) {
#pragma unroll
    for (int dj = 0; dj < 2; ++dj) {
      const int row = ((C0[di] << 7) + C1[dj]) << 7;
      const float W01 = W0c[di] * W1c[dj];
#pragma unroll
      for (int dk = 0; dk < 2; ++dk) {
        const int off = row + C2[dk];             // 32-bit lane offset, shared
        const float Wt = W01 * W2c[dk];
        // Regular-temporal gathers: x stays resident in the 192MB L2;
        // heavy reuse between neighboring lanes (flow is only a few voxels).
        acc0 = fmaf(Wt, xb0[off], acc0);
        acc1 = fmaf(Wt, xb1[off], acc1);
      }
    }
  }
  // Streaming output: non-temporal so it doesn't evict x from L2.
  __builtin_nontemporal_store(acc0, ob0 + v);
  __builtin_nontemporal_store(acc1, ob1 + v);
}

__global__ __launch_bounds__(TPB)
void warp3d_kernel(const float* __restrict__ x, const float* __restrict__ w,
                   float* __restrict__ out) {
  const unsigned lane = threadIdx.x;
  const unsigned bidx = blockIdx.y;                     // scalar batch index
  const unsigned stride = gridDim.x * TPB;              // voxels per sweep
  const unsigned vbase = blockIdx.x * TPB + lane;       // coalesced in v

  // Scalar bases, one per plane/channel, pinned to SGPR pairs so every
  // memory op is saddr + (v << 2) scale_offset with no vector 64-bit adds.
  const float* wp0 = w + (size_t)bidx * 3 * (size_t)VOLN;
  const float* wp1 = wp0 + VOLN;
  const float* wp2 = wp0 + 2 * VOLN;
  const float* xb0 = x + (size_t)bidx * NC * (size_t)VOLN;
  const float* xb1 = xb0 + VOLN;
  float* ob0 = out + (size_t)bidx * NC * (size_t)VOLN;
  float* ob1 = ob0 + VOLN;
  PIN_SGPR(wp0); PIN_SGPR(wp1); PIN_SGPR(wp2);
  PIN_SGPR(xb0); PIN_SGPR(xb1);
  PIN_SGPR(ob0); PIN_SGPR(ob1);

#if USE_ASYNC
  __shared__ float sw[2][3][TPB];                       // 6KB, ping-pong

  // Stage flow components for voxel v into LDS buffer `buf`, lane-private
  // slots (no cross-lane sharing -> only s_wait_asynccnt needed, no barrier).
#define STAGE(buf, v_)                                                       \
  do {                                                                       \
    async_load_b32(wp0 + (v_), &sw[buf][0][lane]);                           \
    async_load_b32(wp1 + (v_), &sw[buf][1][lane]);                           \
    async_load_b32(wp2 + (v_), &sw[buf][2][lane]);                           \
  } while (0)

  STAGE(0, vbase);
#pragma unroll
  for (int it = 0; it < IPT; ++it) {
    const unsigned v = vbase + (unsigned)it * stride;
    if (it + 1 < IPT) {
      STAGE((it + 1) & 1, v + stride);     // prefetch next chunk's flow
      __builtin_amdgcn_s_wait_asynccnt(3); // current chunk's 3 loads landed
    } else {
      __builtin_amdgcn_s_wait_asynccnt(0);
    }
    asm volatile("" ::: "memory");         // keep LDS reads below the wait
    const int buf = it & 1;
    const float w0 = sw[buf][0][lane];
    const float w1 = sw[buf][1][lane];
    const float w2 = sw[buf][2][lane];
    asm volatile("" ::: "memory");         // reads done before buf reuse
    trilerp_item(xb0, xb1, ob0, ob1, (int)v, w0, w1, w2);
  }
#undef STAGE
#else
  // Fallback: direct NT loads + gfx1250 global_prefetch of next chunk.
#pragma unroll
  for (int it = 0; it < IPT; ++it) {
    const unsigned v = vbase + (unsigned)it * stride;
    if (it + 1 < IPT) {
      const unsigned nx = v + stride;
      __builtin_prefetch(wp0 + nx, 0, 0);          // -> global_prefetch_b8
      __builtin_prefetch(wp1 + nx, 0, 0);
      __builtin_prefetch(wp2 + nx, 0, 0);
    }
    const float w0 = __builtin_nontemporal_load(wp0 + v);
    const float w1 = __builtin_nontemporal_load(wp1 + v);
    const float w2 = __builtin_nontemporal_load(wp2 + v);
    trilerp_item(xb0, xb1, ob0, ob1, (int)v, w0, w1, w2);
  }
#endif
}

extern "C" void kernel_launch(void* const* d_in, const int* in_sizes, int n_in,
                              void* d_out, int out_size, void* d_ws,
                              size_t ws_size, hipStream_t stream) {
  (void)in_sizes; (void)n_in; (void)d_ws; (void)ws_size; (void)out_size;
  const float* x = (const float*)d_in[0];
  const float* w = (const float*)d_in[1];
  float* out = (float*)d_out;

  // grid.x covers one volume (IPT voxels/thread), grid.y = batch (scalar b).
  const int bx = VOLN / (TPB * IPT);      // 2048
  dim3 grid(bx, NB, 1);                   // 16,384 blocks of 8 wave32s
  warp3d_kernel<<<grid, TPB, 0, stream>>>(x, w, out);
}